// GraphConvolution_69526930588078
// MI455X (gfx1250) — compile-verified
//
#include <hip/hip_runtime.h>

typedef __attribute__((ext_vector_type(2))) float v2f;
typedef __attribute__((ext_vector_type(8))) float v8f;

#define DF 128  // feature dim (d_in == d_out == 128)

// ---------------- degree / normalization ----------------

__global__ void k_init_deg(float* __restrict__ deg, int N) {
  int i = blockIdx.x * blockDim.x + threadIdx.x;
  if (i < N) deg[i] = 1.0f;  // self-loop weight
}

__global__ void k_edge_deg(const long long* __restrict__ edge, float* __restrict__ deg,
                           int E, int Ns) {
  int e = blockIdx.x * blockDim.x + threadIdx.x;
  if (e < E) {
    int c = (int)edge[(size_t)E + e];  // target index
    unsafeAtomicAdd(&deg[Ns + c], 1.0f);
  }
}

__global__ void k_rsqrt_inplace(float* __restrict__ deg, int N) {
  int i = blockIdx.x * blockDim.x + threadIdx.x;
  if (i < N) deg[i] = rsqrtf(deg[i]);  // deg >= 1 always
}

// ---------------- fp32 WMMA GEMM: h = [x_s; x_t] @ W ----------------
// One wave computes a 16-row x 128-col output stripe.
// A fragment (16x4 f32): lanes 0-15 hold M=lane, VGPR0=K0 VGPR1=K1;
//                        lanes 16-31 hold M=lane-16, VGPR0=K2 VGPR1=K3.
// B fragment (4x16 f32): mirror layout, read from LDS-resident W.
// C/D (16x16 f32): VGPR r -> row m0+r (lanes 0-15) / m0+8+r (lanes 16-31), col = lane&15.

__global__ __launch_bounds__(256) void k_gemm_wmma(const float* __restrict__ xs,
                                                   const float* __restrict__ xt,
                                                   const float* __restrict__ Wg,
                                                   float* __restrict__ h,
                                                   int Ns, int N) {
  __shared__ float Ws[DF * DF];  // 64 KB
  {
    const float4* src = (const float4*)Wg;
    float4* dst = (float4*)Ws;
    for (int i = threadIdx.x; i < DF * DF / 4; i += 256) dst[i] = src[i];
  }
  __syncthreads();

  const int wave = threadIdx.x >> 5;
  const int lane = threadIdx.x & 31;
  const int tile = blockIdx.x * 8 + wave;
  const int ntiles = N >> 4;  // N is a multiple of 16 (100000 = 6250*16)
  if (tile >= ntiles) return;  // wave-uniform: EXEC stays all-1s for WMMA

  const int m0 = tile << 4;
  const int mrow = m0 + (lane & 15);
  const int khalf = (lane >> 4) << 1;  // 0 or 2
  const int colb = lane & 15;

  const float* xrow = (mrow < Ns) ? (xs + (size_t)mrow * DF)
                                  : (xt + (size_t)(mrow - Ns) * DF);

  v8f acc[8] = {};

  for (int k0 = 0; k0 < DF; k0 += 4) {
    v2f a;
    a.x = xrow[k0 + khalf];
    a.y = xrow[k0 + khalf + 1];
#pragma unroll
    for (int n = 0; n < 8; ++n) {
      v2f b;
      b.x = Ws[(k0 + khalf) * DF + n * 16 + colb];
      b.y = Ws[(k0 + khalf + 1) * DF + n * 16 + colb];
      acc[n] = __builtin_amdgcn_wmma_f32_16x16x4_f32(
          false, a, false, b, (short)0, acc[n], false, false);
    }
  }

  const int srow0 = m0 + ((lane >> 4) << 3);
#pragma unroll
  for (int n = 0; n < 8; ++n) {
#pragma unroll
    for (int r = 0; r < 8; ++r) {
      h[(size_t)(srow0 + r) * DF + n * 16 + colb] = acc[n][r];
    }
  }
}

// ---------------- self-loop init: out = h * dinv^2 ----------------

__global__ void k_self_init(const float* __restrict__ h, const float* __restrict__ dinv,
                            float* __restrict__ out, int total) {
  int i = blockIdx.x * blockDim.x + threadIdx.x;
  if (i < total) {
    float d = dinv[i >> 7];
    out[i] = h[i] * d * d;
  }
}

// ---------------- edge scatter: out[c] += h[r] * dinv[r]*dinv[c] ----------------
// block = 1 edge, thread = 1 feature (coalesced gather, native f32 atomics)

__global__ void k_edge_scatter(const long long* __restrict__ edge,
                               const float* __restrict__ h,
                               const float* __restrict__ dinv,
                               float* __restrict__ out, int E, int Ns) {
  int e = blockIdx.x;
  int f = threadIdx.x;
  int r = (int)edge[e];
  int c = Ns + (int)edge[(size_t)E + e];
  float norm = dinv[r] * dinv[c];
  float v = h[(size_t)r * DF + f] * norm;
  unsafeAtomicAdd(&out[(size_t)c * DF + f], v);
}

__global__ void k_relu(float* __restrict__ out, int total) {
  int i = blockIdx.x * blockDim.x + threadIdx.x;
  if (i < total) out[i] = fmaxf(out[i], 0.0f);
}

// ---------------- launch ----------------

extern "C" void kernel_launch(void* const* d_in, const int* in_sizes, int n_in,
                              void* d_out, int out_size, void* d_ws, size_t ws_size,
                              hipStream_t stream) {
  const long long* edge = (const long long*)d_in[0];  // (2, E) int64
  const float* xs = (const float*)d_in[1];            // (N_s, 128)
  const float* xt = (const float*)d_in[2];            // (N_t, 128)
  const float* Wg = (const float*)d_in[3];            // (128, 128)
  float* out = (float*)d_out;                         // (N, 128)

  const int E = in_sizes[0] / 2;
  const int Ns = in_sizes[1] / DF;
  const int Nt = in_sizes[2] / DF;
  const int N = Ns + Nt;
  const int total = N * DF;

  float* h = (float*)d_ws;                 // N*128 floats (51.2 MB)
  float* dinv = h + (size_t)N * DF;        // N floats (deg -> dinv in place)

  k_init_deg<<<(N + 255) / 256, 256, 0, stream>>>(dinv, N);
  k_edge_deg<<<(E + 255) / 256, 256, 0, stream>>>(edge, dinv, E, Ns);
  k_rsqrt_inplace<<<(N + 255) / 256, 256, 0, stream>>>(dinv, N);

  const int ntiles = N >> 4;                // 6250
  const int gblocks = (ntiles + 7) / 8;     // 782 blocks x 8 waves
  k_gemm_wmma<<<gblocks, 256, 0, stream>>>(xs, xt, Wg, h, Ns, N);

  k_self_init<<<(total + 255) / 256, 256, 0, stream>>>(h, dinv, out, total);
  k_edge_scatter<<<E, DF, 0, stream>>>(edge, h, dinv, out, E, Ns);
  k_relu<<<(total + 255) / 256, 256, 0, stream>>>(out, total);
}